// UTDAttnHeadNet_6176162972411
// MI455X (gfx1250) — compile-verified
//
#include <hip/hip_runtime.h>
#include <hip/hip_bf16.h>
#include <math.h>
#include <stdint.h>

typedef float v2f __attribute__((ext_vector_type(2)));
typedef float v8f __attribute__((ext_vector_type(8)));

#define N_TOK   8192
#define D_IN    384
#define D_H     64
#define N_HEADS 4
#define N_CLS   6
#define LN_EPS  1e-5f

// workspace layout (float offsets)
#define WS_H    0
#define WS_HP   (WS_H   + N_TOK * D_H)                 // [H][N][64]
#define WS_SSQ  (WS_HP  + N_HEADS * N_TOK * D_H)       // [H][N]
#define WS_HOUT (WS_SSQ + N_HEADS * N_TOK)             // [H][N][64]

// ---------------------------------------------------------------------------
// CDNA5 async global -> LDS copy (ASYNCcnt path), one 16B packet per lane.
// ---------------------------------------------------------------------------
__device__ __forceinline__ void async_copy_b128(uint32_t lds_byte_addr,
                                                const float* gptr)
{
    asm volatile("global_load_async_to_lds_b128 %0, %1, off"
                 :: "v"(lds_byte_addr),
                    "v"((unsigned long long)(uintptr_t)gptr)
                 : "memory");
}
__device__ __forceinline__ void wait_asynccnt0()
{
    asm volatile("s_wait_asynccnt 0x0" ::: "memory");
}

// ---------------------------------------------------------------------------
// h = relu(x @ proj_w + proj_b)   [N,64]   via V_WMMA_F32_16X16X4_F32
// one wave per 16-row tile; A = x rows, B = proj_w, C seeded with bias.
// ---------------------------------------------------------------------------
__global__ __launch_bounds__(256) void k_proj(const float* __restrict__ x,
                                              const float* __restrict__ w,
                                              const float* __restrict__ b,
                                              float* __restrict__ h)
{
    const int wave = threadIdx.x >> 5;
    const int lane = threadIdx.x & 31;
    const int hi   = lane >> 4;
    const int lj   = lane & 15;
    const int n0   = (blockIdx.x * 8 + wave) << 4;    // 512 waves total

    v8f acc[4];
#pragma unroll
    for (int cb = 0; cb < 4; ++cb) {
        const float bb = b[16 * cb + lj];
#pragma unroll
        for (int r = 0; r < 8; ++r) acc[cb][r] = bb;
    }

#pragma unroll 4
    for (int kc = 0; kc < D_IN / 4; ++kc) {           // 96 K-chunks
        v2f a = *(const v2f*)&x[(n0 + lj) * D_IN + 4 * kc + 2 * hi];
#pragma unroll
        for (int cb = 0; cb < 4; ++cb) {
            v2f bv;
            bv.x = w[(4 * kc + 2 * hi + 0) * D_H + 16 * cb + lj];
            bv.y = w[(4 * kc + 2 * hi + 1) * D_H + 16 * cb + lj];
            acc[cb] = __builtin_amdgcn_wmma_f32_16x16x4_f32(false, a, false, bv,
                                                            (short)0, acc[cb],
                                                            false, false);
        }
    }

#pragma unroll
    for (int cb = 0; cb < 4; ++cb)
#pragma unroll
        for (int r = 0; r < 8; ++r)
            h[(n0 + r + 8 * hi) * D_H + 16 * cb + lj] = fmaxf(acc[cb][r], 0.0f);
}

// ---------------------------------------------------------------------------
// hp[h] = h @ head_w[h] + head_b[h]   [H,N,64]   + fused ssq[h][n]
// ---------------------------------------------------------------------------
__global__ __launch_bounds__(256) void k_heads(const float* __restrict__ h,
                                               const float* __restrict__ head_w,
                                               const float* __restrict__ head_b,
                                               float* __restrict__ hp,
                                               float* __restrict__ ssq)
{
    const int wave = threadIdx.x >> 5;
    const int lane = threadIdx.x & 31;
    const int hi   = lane >> 4;
    const int lj   = lane & 15;
    const int gw   = blockIdx.x * 8 + wave;           // 2048 waves
    const int hd   = gw >> 9;
    const int n0   = (gw & 511) << 4;
    const float* w = head_w + hd * D_H * D_H;

    v8f acc[4];
#pragma unroll
    for (int cb = 0; cb < 4; ++cb) {
        const float bb = head_b[hd * D_H + 16 * cb + lj];
#pragma unroll
        for (int r = 0; r < 8; ++r) acc[cb][r] = bb;
    }

#pragma unroll
    for (int kc = 0; kc < 16; ++kc) {
        v2f a = *(const v2f*)&h[(n0 + lj) * D_H + 4 * kc + 2 * hi];
#pragma unroll
        for (int cb = 0; cb < 4; ++cb) {
            v2f bv;
            bv.x = w[(4 * kc + 2 * hi + 0) * D_H + 16 * cb + lj];
            bv.y = w[(4 * kc + 2 * hi + 1) * D_H + 16 * cb + lj];
            acc[cb] = __builtin_amdgcn_wmma_f32_16x16x4_f32(false, a, false, bv,
                                                            (short)0, acc[cb],
                                                            false, false);
        }
    }

    float* dst = hp + hd * (N_TOK * D_H);
    float sums[8];
#pragma unroll
    for (int r = 0; r < 8; ++r) sums[r] = 0.0f;
#pragma unroll
    for (int cb = 0; cb < 4; ++cb)
#pragma unroll
        for (int r = 0; r < 8; ++r) {
            const float v = acc[cb][r];
            dst[(n0 + r + 8 * hi) * D_H + 16 * cb + lj] = v;
            sums[r] += v * v;
        }
    // reduce over the 16-lane column group (rows are r + 8*hi)
#pragma unroll
    for (int off = 1; off < 16; off <<= 1)
#pragma unroll
        for (int r = 0; r < 8; ++r)
            sums[r] += __shfl_xor(sums[r], off, 32);
    if (lj == 0) {
#pragma unroll
        for (int r = 0; r < 8; ++r)
            ssq[hd * N_TOK + n0 + r + 8 * hi] = sums[r];
    }
}

// ---------------------------------------------------------------------------
// Streaming distance-softmax attention, per head.
// One wave <-> one (head, 16-query tile).  Block = 8 waves, all same head.
// Key tiles double-buffered in LDS via async global->LDS loads (ASYNCcnt).
// S^T = Kt * Q^T (16 wmma), online softmax, acc += P * Kt (16 wmma).
// D layout (ISA 7.12.2): lane L, vgpr r -> row = r + 8*(L>=16), col = L&15.
// ---------------------------------------------------------------------------
__global__ __launch_bounds__(256) void k_attn(const float* __restrict__ hp,
                                              const float* __restrict__ ssq,
                                              float* __restrict__ hout)
{
    __shared__ __align__(16) float ktile[2][16 * D_H];  // double-buffered key tile
    __shared__ float ptile[8][256];                     // per-wave P tile (16x16)
    __shared__ float sctile[8][16];                     // per-wave transpose buffer

    const int tid  = threadIdx.x;
    const int wave = tid >> 5;
    const int lane = tid & 31;
    const int hi   = lane >> 4;
    const int lj   = lane & 15;

    const int gw   = blockIdx.x * 8 + wave;  // 2048 waves = 4 heads * 512 tiles
    const int head = gw >> 9;
    const int n0   = (gw & 511) << 4;

    const float* hph  = hp  + head * (N_TOK * D_H);
    const float* ssqh = ssq + head * N_TOK;

    const uint32_t lds_k[2] = {
        (uint32_t)(uintptr_t)(&ktile[0][0]) + (uint32_t)tid * 16u,
        (uint32_t)(uintptr_t)(&ktile[1][0]) + (uint32_t)tid * 16u };

    // B operand of S^T gemm: hp_q rows.  b[v] = hp_q[lj][4*kc + 2*hi + v]
    v2f qb[16];
#pragma unroll
    for (int kc = 0; kc < 16; ++kc)
        qb[kc] = *(const v2f*)&hph[(n0 + lj) * D_H + 4 * kc + 2 * hi];
    const float ssqq = ssqh[n0 + lj];

    float run_m = -INFINITY;
    float run_l = 0.0f;
    v8f acc[4] = {};

    // prefetch tile 0 into buffer 0
    async_copy_b128(lds_k[0], hph + tid * 4);
    wait_asynccnt0();
    __syncthreads();

    const int NT = N_TOK / 16;
    for (int t = 0; t < NT; ++t) {
        const int m0 = t << 4;
        const float* kt = ktile[t & 1];

        // prefetch next tile into the other buffer (readers finished last barrier)
        if (t + 1 < NT)
            async_copy_b128(lds_k[(t + 1) & 1],
                            hph + (m0 + 16) * D_H + tid * 4);

        // S^T tile: rows = keys (i = r + 8*hi), cols = queries (j = lj)
        v8f c = {};
#pragma unroll
        for (int kc = 0; kc < 16; ++kc) {
            v2f a = *(const v2f*)&kt[lj * D_H + 4 * kc + 2 * hi];
            c = __builtin_amdgcn_wmma_f32_16x16x4_f32(false, a, false, qb[kc],
                                                      (short)0, c, false, false);
        }

        // scores: s = -sqrt(max(ssq_i + ssq_j - 2*dot, 0))
        float pr[8];
        float tmax = -INFINITY;
#pragma unroll
        for (int r = 0; r < 8; ++r) {
            float sqd = ssqh[m0 + r + 8 * hi] + ssqq - 2.0f * c[r];
            sqd = fmaxf(sqd, 0.0f);
            pr[r] = -sqrtf(sqd);
            tmax = fmaxf(tmax, pr[r]);
        }
        tmax = fmaxf(tmax, __shfl_xor(tmax, 16, 32));
        const float nm   = fmaxf(run_m, tmax);
        const float corr = __expf(run_m - nm);     // 0 on first tile (run_m=-inf)
        float tsum = 0.0f;
#pragma unroll
        for (int r = 0; r < 8; ++r) {
            const float p = __expf(pr[r] - nm);
            tsum += p;
            ptile[wave][lj * 16 + r + 8 * hi] = p; // store P[n=lj][m=r+8*hi]
        }
        tsum += __shfl_xor(tsum, 16, 32);
        run_l = run_l * corr + tsum;
        run_m = nm;
        if (lane < 16) sctile[wave][lane] = corr;  // per-query correction

        // rescale acc: acc rows are queries n = r + 8*hi
        float sc[8];
#pragma unroll
        for (int r = 0; r < 8; ++r) sc[r] = sctile[wave][r + 8 * hi];
#pragma unroll
        for (int cb = 0; cb < 4; ++cb)
#pragma unroll
            for (int r = 0; r < 8; ++r) acc[cb][r] *= sc[r];

        // acc[n][e] += P[n][m] * hp_k[m][e]
#pragma unroll
        for (int cb = 0; cb < 4; ++cb) {
#pragma unroll
            for (int kc = 0; kc < 4; ++kc) {
                v2f a = *(const v2f*)&ptile[wave][lj * 16 + 4 * kc + 2 * hi];
                v2f bv;
                bv.x = kt[(4 * kc + 2 * hi + 0) * D_H + 16 * cb + lj];
                bv.y = kt[(4 * kc + 2 * hi + 1) * D_H + 16 * cb + lj];
                acc[cb] = __builtin_amdgcn_wmma_f32_16x16x4_f32(false, a, false, bv,
                                                                (short)0, acc[cb],
                                                                false, false);
            }
        }

        wait_asynccnt0();       // next tile landed in LDS
        __syncthreads();        // everyone done reading current tile
    }

    // normalize by running sum (transpose lane-indexed l to row-indexed)
    if (lane < 16) sctile[wave][lane] = run_l;
    float linv[8];
#pragma unroll
    for (int r = 0; r < 8; ++r) linv[r] = 1.0f / sctile[wave][r + 8 * hi];

    float* dst = hout + head * (N_TOK * D_H);
#pragma unroll
    for (int cb = 0; cb < 4; ++cb)
#pragma unroll
        for (int r = 0; r < 8; ++r)
            dst[(n0 + r + 8 * hi) * D_H + 16 * cb + lj] = acc[cb][r] * linv[r];
}

// ---------------------------------------------------------------------------
// combine heads (softmax(attn_w)), LayerNorm, FC head.   One thread per row.
// ---------------------------------------------------------------------------
__global__ __launch_bounds__(256) void k_final(const float* __restrict__ hout,
                                               const float* __restrict__ attn_w,
                                               const float* __restrict__ gamma,
                                               const float* __restrict__ beta,
                                               const float* __restrict__ fc_w,
                                               const float* __restrict__ fc_b,
                                               float* __restrict__ out)
{
    const int n = blockIdx.x * 256 + threadIdx.x;
    if (n >= N_TOK) return;

    float aw[N_HEADS];
    float amax = -INFINITY;
#pragma unroll
    for (int h = 0; h < N_HEADS; ++h) { aw[h] = attn_w[h]; amax = fmaxf(amax, aw[h]); }
    float asum = 0.0f;
#pragma unroll
    for (int h = 0; h < N_HEADS; ++h) { aw[h] = __expf(aw[h] - amax); asum += aw[h]; }
    const float ainv = 1.0f / asum;

    float comb[D_H];
#pragma unroll
    for (int e = 0; e < D_H; ++e) comb[e] = 0.0f;
#pragma unroll
    for (int h = 0; h < N_HEADS; ++h) {
        const float* p = hout + h * (N_TOK * D_H) + n * D_H;
        const float w = aw[h] * ainv;
        for (int e = 0; e < D_H; ++e) comb[e] += w * p[e];
    }

    float mu = 0.0f;
#pragma unroll
    for (int e = 0; e < D_H; ++e) mu += comb[e];
    mu *= (1.0f / D_H);
    float var = 0.0f;
#pragma unroll
    for (int e = 0; e < D_H; ++e) { const float d = comb[e] - mu; var += d * d; }
    var *= (1.0f / D_H);
    const float rs = rsqrtf(var + LN_EPS);
#pragma unroll
    for (int e = 0; e < D_H; ++e)
        comb[e] = (comb[e] - mu) * rs * gamma[e] + beta[e];

#pragma unroll
    for (int c = 0; c < N_CLS; ++c) {
        float a = fc_b[c];
        for (int e = 0; e < D_H; ++e) a += comb[e] * fc_w[e * N_CLS + c];
        out[n * N_CLS + c] = a;
    }
}

// ---------------------------------------------------------------------------
extern "C" void kernel_launch(void* const* d_in, const int* in_sizes, int n_in,
                              void* d_out, int out_size, void* d_ws, size_t ws_size,
                              hipStream_t stream)
{
    const float* x      = (const float*)d_in[0];
    const float* proj_w = (const float*)d_in[1];
    const float* proj_b = (const float*)d_in[2];
    const float* head_w = (const float*)d_in[3];
    const float* head_b = (const float*)d_in[4];
    const float* attn_w = (const float*)d_in[5];
    const float* gamma  = (const float*)d_in[6];
    const float* beta   = (const float*)d_in[7];
    const float* fc_w   = (const float*)d_in[8];
    const float* fc_b   = (const float*)d_in[9];

    float* ws   = (float*)d_ws;
    float* h    = ws + WS_H;
    float* hp   = ws + WS_HP;
    float* ssq  = ws + WS_SSQ;
    float* hout = ws + WS_HOUT;
    float* out  = (float*)d_out;

    k_proj <<<(N_TOK / 16) / 8,             256, 0, stream>>>(x, proj_w, proj_b, h);
    k_heads<<<(N_HEADS * N_TOK / 16) / 8,   256, 0, stream>>>(h, head_w, head_b, hp, ssq);
    k_attn <<<(N_HEADS * (N_TOK / 16)) / 8, 256, 0, stream>>>(hp, ssq, hout);
    k_final<<<N_TOK / 256,                  256, 0, stream>>>(hout, attn_w, gamma, beta,
                                                              fc_w, fc_b, out);
}